// PositionalGATModel_7748121002024
// MI455X (gfx1250) — compile-verified
//
#include <hip/hip_runtime.h>
#include <math.h>

#define GG 100
#define HID 128           // HEADS*CH
#define IN_DIM 128
#define POSD 16
#define OUT_DIM 64
#define NEG_SLOPE 0.2f
#define BN_EPS 1e-5f
#define KMAX 144
#define LDS_PAD 4         // stride padding: conflict-free banks for K=128 and K=144

typedef __attribute__((ext_vector_type(2))) float v2f;
typedef __attribute__((ext_vector_type(8))) float v8f;

// ---------------- helpers ----------------
__device__ __forceinline__ void atomicMaxF(float* addr, float val) {
    // Ordered int/uint trick: works for any sign mix; init pattern may be
    // 0xFFFFFFFF (NaN) which loses to everything under both orderings.
    if (val >= 0.0f) atomicMax((int*)addr, __float_as_int(val));
    else             atomicMin((unsigned int*)addr, __float_as_uint(val));
}

// ---------------- setup: positions + concat ----------------
__global__ void k_count(const int* __restrict__ batch, int* __restrict__ counts, int n) {
    int i = blockIdx.x * blockDim.x + threadIdx.x;
    if (i < n) atomicAdd(&counts[batch[i]], 1);
}

__global__ void k_starts(const int* __restrict__ counts, int* __restrict__ starts) {
    if (threadIdx.x == 0) {
        int acc = 0;
        for (int g = 0; g < GG; ++g) { starts[g] = acc; acc += counts[g]; }
    }
}

__global__ void k_pos(const int* __restrict__ batch, const int* __restrict__ counts,
                      const int* __restrict__ starts, float* __restrict__ pos, int n) {
    int i = blockIdx.x * blockDim.x + threadIdx.x;
    if (i >= n) return;
    int b = batch[i];
    int local = i - starts[b];
    int cnt = counts[b];
    int gs = (int)ceilf(sqrtf((float)cnt));     // same float math as reference
    int row = local / gs;
    int col = local - row * gs;
    float denom = (float)((gs - 1) > 1 ? (gs - 1) : 1);
    pos[i * 2 + 0] = (float)row / denom;
    pos[i * 2 + 1] = (float)col / denom;
}

__global__ void k_concat(const float* __restrict__ x, const float* __restrict__ pos,
                         const float* __restrict__ peW, const float* __restrict__ peb,
                         float* __restrict__ h, int n) {
    int idx = blockIdx.x * blockDim.x + threadIdx.x;
    int total = n * (IN_DIM + POSD);
    if (idx >= total) return;
    int i = idx / (IN_DIM + POSD);
    int j = idx - i * (IN_DIM + POSD);
    if (j < IN_DIM) {
        h[idx] = x[i * IN_DIM + j];
    } else {
        int p = j - IN_DIM;
        h[idx] = pos[i * 2] * peW[p * 2] + pos[i * 2 + 1] * peW[p * 2 + 1] + peb[p];
    }
}

// ---------------- fp32 WMMA GEMM: C[M,128] = A[M,K] @ W[128,K]^T ----------------
// Block = 256 threads = 8 waves; wave w owns output cols [w*16, w*16+16);
// blockIdx.x owns output rows [bx*16, bx*16+16). M divisible by 16, K by 4.
// The 16xK A tile is staged once per block into LDS (coalesced global_load_b32),
// then all 8 waves read their WMMA fragments from LDS (ds_load, padded stride
// -> bank-conflict-free for K=128 and K=144). W (73KB) stays in global: it is
// shared by all blocks and lives in L0/L2 after first touch.
__global__ __launch_bounds__(256) void k_gemm_wmma(const float* __restrict__ A,
                                                   const float* __restrict__ W,
                                                   float* __restrict__ C, int K) {
    __shared__ float aTile[16 * (KMAX + LDS_PAD)];
    const int stride = K + LDS_PAD;
    const int mBase = blockIdx.x * 16;

    // cooperative, coalesced stage of A[mBase:mBase+16, 0:K] into LDS
    for (int i = threadIdx.x; i < 16 * K; i += 256) {
        int r = i / K;
        int c = i - r * K;
        aTile[r * stride + c] = A[(size_t)(mBase + r) * K + c];
    }
    __syncthreads();

    int wave = threadIdx.x >> 5;
    int lane = threadIdx.x & 31;
    int nBase = wave * 16;
    int l16  = lane & 15;
    int kOff = (lane >> 4) * 2;                 // 0 or 2: K pair owned by this half-wave

    const float* aRow = aTile + l16 * stride;            // A fragment row (M = lane%16)
    const float* bCol = W + (size_t)(nBase + l16) * K;   // B^T row == W row (N = lane%16)

    v8f acc = {};
    for (int k = 0; k < K; k += 4) {
        v2f a, b;
        a.x = aRow[k + kOff];     a.y = aRow[k + kOff + 1];
        b.x = bCol[k + kOff];     b.y = bCol[k + kOff + 1];
        acc = __builtin_amdgcn_wmma_f32_16x16x4_f32(false, a, false, b,
                                                    (short)0, acc, false, false);
    }
    // C/D layout: VGPR r holds row mBase + (lane/16)*8 + r, col nBase + lane%16
    int rowBase = mBase + (lane >> 4) * 8;
    int col = nBase + l16;
#pragma unroll
    for (int r = 0; r < 8; ++r)
        C[(size_t)(rowBase + r) * HID + col] = acc[r];
}

// ---------------- per-node attention dots ----------------
__global__ void k_adot(const float* __restrict__ xt, const float* __restrict__ att,
                       float* __restrict__ asrc, float* __restrict__ adst, int n) {
    int tid = blockIdx.x * blockDim.x + threadIdx.x;
    if (tid >= n * 4) return;
    int node = tid >> 2;
    int h = tid & 3;
    const float* xp = xt + (size_t)node * HID + h * 32;
    const float* ap = att + h * 80;
    float s1 = 0.f, s2 = 0.f;
#pragma unroll
    for (int c = 0; c < 32; ++c) { s1 += xp[c] * ap[c]; s2 += xp[c] * ap[32 + c]; }
    asrc[tid] = s1;
    adst[tid] = s2;
}

// q[h][d] = sum_p att[h, 64+p] * posW[p, d]   (4x2)
__global__ void k_qvec(const float* __restrict__ att, const float* __restrict__ posW,
                       float* __restrict__ q) {
    int tid = threadIdx.x;
    if (tid >= 8) return;
    int h = tid >> 1, d = tid & 1;
    float s = 0.f;
    for (int p = 0; p < POSD; ++p) s += att[h * 80 + 64 + p] * posW[p * 2 + d];
    q[tid] = s;
}

// ---------------- edge passes ----------------
__global__ void k_edge_max(const int* __restrict__ src, const int* __restrict__ dst,
                           int E, int n, const float* __restrict__ pos,
                           const float* __restrict__ asrc, const float* __restrict__ adst,
                           const float* __restrict__ q, float* __restrict__ amax) {
    int e = blockIdx.x * blockDim.x + threadIdx.x;
    if (e >= E + n) return;
    int s, d; float dx, dy;
    if (e < E) {
        s = src[e]; d = dst[e];
        dx = pos[d * 2] - pos[s * 2];
        dy = pos[d * 2 + 1] - pos[s * 2 + 1];
    } else { s = d = e - E; dx = 0.f; dy = 0.f; }   // self loop
#pragma unroll
    for (int h = 0; h < 4; ++h) {
        float a = asrc[s * 4 + h] + adst[d * 4 + h] + dx * q[h * 2] + dy * q[h * 2 + 1];
        a = a > 0.f ? a : NEG_SLOPE * a;
        atomicMaxF(&amax[d * 4 + h], a);
    }
}

__global__ void k_fixmax(float* __restrict__ amax, int n4) {
    int i = blockIdx.x * blockDim.x + threadIdx.x;
    if (i >= n4) return;
    float v = amax[i];
    if (!__builtin_isfinite(v)) amax[i] = 0.f;
}

__global__ void k_edge_sum(const int* __restrict__ src, const int* __restrict__ dst,
                           int E, int n, const float* __restrict__ pos,
                           const float* __restrict__ asrc, const float* __restrict__ adst,
                           const float* __restrict__ q, const float* __restrict__ amax,
                           float* __restrict__ ssum) {
    int e = blockIdx.x * blockDim.x + threadIdx.x;
    if (e >= E + n) return;
    int s, d; float dx, dy;
    if (e < E) {
        s = src[e]; d = dst[e];
        dx = pos[d * 2] - pos[s * 2];
        dy = pos[d * 2 + 1] - pos[s * 2 + 1];
    } else { s = d = e - E; dx = 0.f; dy = 0.f; }
#pragma unroll
    for (int h = 0; h < 4; ++h) {
        float a = asrc[s * 4 + h] + adst[d * 4 + h] + dx * q[h * 2] + dy * q[h * 2 + 1];
        a = a > 0.f ? a : NEG_SLOPE * a;
        atomicAdd(&ssum[d * 4 + h], expf(a - amax[d * 4 + h]));
    }
}

// out[n,c] = xt[n,c] * s/(s+1e-16) + bias[c]   (softmax-weighted self aggregation)
__global__ void k_finalize(const float* __restrict__ xt, const float* __restrict__ ssum,
                           const float* __restrict__ bias, float* __restrict__ out, int n) {
    int idx = blockIdx.x * blockDim.x + threadIdx.x;
    if (idx >= n * HID) return;
    int node = idx >> 7;
    int c = idx & 127;
    int h = c >> 5;
    float s = ssum[node * 4 + h];
    out[idx] = xt[idx] * (s / (s + 1e-16f)) + bias[c];
}

// ---------------- batchnorm (biased var) + ELU ----------------
__global__ void k_bn_stats(const float* __restrict__ h, float* __restrict__ sums, int n) {
    int c = threadIdx.x;        // blockDim = 128
    float s = 0.f, s2 = 0.f;
    for (int r = blockIdx.x; r < n; r += gridDim.x) {
        float v = h[(size_t)r * HID + c];
        s += v; s2 += v * v;
    }
    atomicAdd(&sums[c], s);
    atomicAdd(&sums[HID + c], s2);
}

__global__ void k_bn_apply(float* __restrict__ h, const float* __restrict__ sums,
                           const float* __restrict__ g, const float* __restrict__ b, int n) {
    int idx = blockIdx.x * blockDim.x + threadIdx.x;
    if (idx >= n * HID) return;
    int c = idx & 127;
    float inv_n = 1.f / (float)n;
    float mu = sums[c] * inv_n;
    float var = sums[HID + c] * inv_n - mu * mu;
    float y = (h[idx] - mu) * rsqrtf(var + BN_EPS) * g[c] + b[c];
    h[idx] = y > 0.f ? y : expm1f(y);           // ELU
}

// ---------------- pooling + FC ----------------
__global__ void k_pool(const float* __restrict__ h, const int* __restrict__ counts,
                       const int* __restrict__ starts, float* __restrict__ pooled) {
    int g = blockIdx.x;         // GG blocks
    int c = threadIdx.x;        // 128 threads
    int cnt = counts[g], st = starts[g];
    float s = 0.f;
    for (int i = 0; i < cnt; ++i) s += h[(size_t)(st + i) * HID + c];
    pooled[g * HID + c] = s / (float)cnt;
}

__global__ void k_fc(const float* __restrict__ pooled, const float* __restrict__ W,
                     const float* __restrict__ b, float* __restrict__ out) {
    int g = blockIdx.x;         // GG blocks
    int o = threadIdx.x;        // 64 threads
    float s = b[o];
    for (int c = 0; c < HID; ++c) s += pooled[g * HID + c] * W[o * HID + c];
    out[g * OUT_DIM + o] = s;
}

// ---------------- host side ----------------
extern "C" void kernel_launch(void* const* d_in, const int* in_sizes, int n_in,
                              void* d_out, int out_size, void* d_ws, size_t ws_size,
                              hipStream_t stream) {
    const float* x     = (const float*)d_in[0];
    const int*   eidx  = (const int*)  d_in[1];
    const int*   batch = (const int*)  d_in[2];
    const float* peW   = (const float*)d_in[3];
    const float* peb   = (const float*)d_in[4];
    const float* linW[2]  = {(const float*)d_in[5],  (const float*)d_in[11]};
    const float* att[2]   = {(const float*)d_in[6],  (const float*)d_in[12]};
    const float* posW[2]  = {(const float*)d_in[7],  (const float*)d_in[13]};
    const float* bias[2]  = {(const float*)d_in[8],  (const float*)d_in[14]};
    const float* bng[2]   = {(const float*)d_in[9],  (const float*)d_in[15]};
    const float* bnb[2]   = {(const float*)d_in[10], (const float*)d_in[16]};
    const float* fcW   = (const float*)d_in[17];
    const float* fcb   = (const float*)d_in[18];
    float* out = (float*)d_out;

    const int N = in_sizes[0] / IN_DIM;      // 100000
    const int E = in_sizes[1] / 2;           // 1600000
    const int* src = eidx;
    const int* dst = eidx + E;

    // workspace carve-up (fp32)
    float* f      = (float*)d_ws;
    float* pos    = f;                               // 2N
    float* hbuf   = pos   + (size_t)2 * N;           // 144N (reused as 128-wide layer output)
    float* xt     = hbuf  + (size_t)(IN_DIM+POSD)*N; // 128N
    float* asrc   = xt    + (size_t)HID * N;         // 4N
    float* adst   = asrc  + (size_t)4 * N;           // 4N
    float* amax   = adst  + (size_t)4 * N;           // 4N
    float* ssum   = amax  + (size_t)4 * N;           // 4N
    float* bns    = ssum  + (size_t)4 * N;           // 256
    float* qv     = bns   + 256;                     // 8
    float* pooled = qv    + 8;                       // GG*128
    int*   counts = (int*)(pooled + GG * HID);       // GG
    int*   starts = counts + GG;                     // GG

    const int TB = 256;
    int nB       = (N + TB - 1) / TB;
    int concatB  = (N * (IN_DIM + POSD) + TB - 1) / TB;
    int edgeB    = (E + N + TB - 1) / TB;
    int n4B      = (N * 4 + TB - 1) / TB;
    int nhB      = (N * HID + TB - 1) / TB;

    // ---- positions + input concat ----
    hipMemsetAsync(counts, 0, GG * sizeof(int), stream);
    k_count <<<nB, TB, 0, stream>>>(batch, counts, N);
    k_starts<<<1, 32, 0, stream>>>(counts, starts);
    k_pos   <<<nB, TB, 0, stream>>>(batch, counts, starts, pos, N);
    k_concat<<<concatB, TB, 0, stream>>>(x, pos, peW, peb, hbuf, N);

    // ---- two GAT layers ----
    for (int l = 0; l < 2; ++l) {
        int K = (l == 0) ? (IN_DIM + POSD) : HID;
        k_gemm_wmma<<<N / 16, 256, 0, stream>>>(hbuf, linW[l], xt, K);
        k_adot<<<n4B, TB, 0, stream>>>(xt, att[l], asrc, adst, N);
        k_qvec<<<1, 32, 0, stream>>>(att[l], posW[l], qv);

        hipMemsetAsync(amax, 0xFF, (size_t)4 * N * sizeof(float), stream); // NaN sentinel
        hipMemsetAsync(ssum, 0,    (size_t)4 * N * sizeof(float), stream);
        k_edge_max<<<edgeB, TB, 0, stream>>>(src, dst, E, N, pos, asrc, adst, qv, amax);
        k_fixmax  <<<n4B, TB, 0, stream>>>(amax, N * 4);
        k_edge_sum<<<edgeB, TB, 0, stream>>>(src, dst, E, N, pos, asrc, adst, qv, amax, ssum);

        k_finalize<<<nhB, TB, 0, stream>>>(xt, ssum, bias[l], hbuf, N);  // hbuf now [N,128]

        hipMemsetAsync(bns, 0, 2 * HID * sizeof(float), stream);
        k_bn_stats<<<512, HID, 0, stream>>>(hbuf, bns, N);
        k_bn_apply<<<nhB, TB, 0, stream>>>(hbuf, bns, bng[l], bnb[l], N);
    }

    // ---- pool + FC ----
    k_pool<<<GG, HID, 0, stream>>>(hbuf, counts, starts, pooled);
    k_fc  <<<GG, OUT_DIM, 0, stream>>>(pooled, fcW, fcb, out);
}